// SpatialGCNEncoder_34540126994670
// MI455X (gfx1250) — compile-verified
//
#include <hip/hip_runtime.h>

typedef float v2f __attribute__((ext_vector_type(2)));
typedef float v8f __attribute__((ext_vector_type(8)));

#define HIDC 64
#define BN_EPS 1e-5f

// ---------------------------------------------------------------------------
// WMMA f32 GEMM: C[N,64] = A[N,K] @ W[K,64] (+ bias per column, optional).
// One wave per 16-row tile; 4 accumulators cover the 64 output columns.
// V_WMMA_F32_16X16X4_F32 layouts (ISA 7.12.2):
//   A 16x4 : lane L holds row M=L&15; VGPR0/1 = K = 2*(L>>4) + {0,1}
//   B 4x16 : VGPR r holds row K = r + 2*(L>>4), col N = L&15
//   C 16x16: VGPR r holds row M = r + 8*(L>>4), col N = L&15
// ---------------------------------------------------------------------------
__global__ void gcn_gemm_wmma(const float* __restrict__ A,
                              const float* __restrict__ W,
                              const float* __restrict__ bias,
                              float* __restrict__ C,
                              int N, int K) {
  const int wave = (int)((blockIdx.x * blockDim.x + threadIdx.x) >> 5);
  const int lane = (int)(threadIdx.x & 31);
  const int row0 = wave << 4;
  if (row0 >= N) return;                    // wave-uniform: EXEC stays all-ones
  const int m    = lane & 15;
  const int half = lane >> 4;               // 0 or 1
  const float* arow = A + (size_t)(row0 + m) * K;

  v8f acc0 = {}, acc1 = {}, acc2 = {}, acc3 = {};
  for (int k = 0; k < K; k += 4) {
    const int ka = k + (half << 1);
    v2f a;
    a.x = arow[ka];
    a.y = arow[ka + 1];
    const float* wr0 = W + (size_t)ka * HIDC + m;   // row K=ka, col base m
    const float* wr1 = wr0 + HIDC;                  // row K=ka+1
    v2f b0, b1, b2, b3;
    b0.x = wr0[0];  b0.y = wr1[0];
    b1.x = wr0[16]; b1.y = wr1[16];
    b2.x = wr0[32]; b2.y = wr1[32];
    b3.x = wr0[48]; b3.y = wr1[48];
    acc0 = __builtin_amdgcn_wmma_f32_16x16x4_f32(false, a, false, b0, (short)0, acc0, false, false);
    acc1 = __builtin_amdgcn_wmma_f32_16x16x4_f32(false, a, false, b1, (short)0, acc1, false, false);
    acc2 = __builtin_amdgcn_wmma_f32_16x16x4_f32(false, a, false, b2, (short)0, acc2, false, false);
    acc3 = __builtin_amdgcn_wmma_f32_16x16x4_f32(false, a, false, b3, (short)0, acc3, false, false);
  }

  float bv0 = 0.f, bv1 = 0.f, bv2 = 0.f, bv3 = 0.f;
  if (bias) { bv0 = bias[m]; bv1 = bias[16 + m]; bv2 = bias[32 + m]; bv3 = bias[48 + m]; }

#pragma unroll
  for (int r = 0; r < 8; ++r) {
    const int grow = row0 + r + (half << 3);
    if (grow < N) {
      float* crow = C + (size_t)grow * HIDC + m;
      crow[0]  = acc0[r] + bv0;
      crow[16] = acc1[r] + bv1;
      crow[32] = acc2[r] + bv2;
      crow[48] = acc3[r] + bv3;
    }
  }
}

// deg[col[e]] += w[e]  (self-loop +1 folded into dis kernel)
__global__ void gcn_deg(const int* __restrict__ col, const float* __restrict__ w,
                        float* __restrict__ deg, int E) {
  int e = (int)(blockIdx.x * blockDim.x + threadIdx.x);
  if (e < E) atomicAdd(&deg[col[e]], w[e]);
}

// dis[i] = d > 0 ? rsqrt(d) : 0, with d = deg_edges[i] + 1 (self-loop weight 1)
__global__ void gcn_dis(float* __restrict__ dis, int N) {
  int i = (int)(blockIdx.x * blockDim.x + threadIdx.x);
  if (i < N) {
    float d = dis[i] + 1.0f;
    dis[i] = (d > 0.0f) ? rsqrtf(d) : 0.0f;
  }
}

// agg[i,c] = bias[c] + hw[i,c] * dis[i]^2   (bias + self-loop contribution)
__global__ void gcn_agg_init(const float* __restrict__ hw, const float* __restrict__ dis,
                             const float* __restrict__ bias, float* __restrict__ agg,
                             int N) {
  int idx = (int)(blockIdx.x * blockDim.x + threadIdx.x);
  if (idx < N * HIDC) {
    int i = idx >> 6;
    int c = idx & 63;
    float di = dis[i];
    agg[idx] = bias[c] + hw[idx] * di * di;
  }
}

// 16 threads per edge, 4 channels each: agg[col] += hw[row] * norm
__global__ void gcn_scatter(const int* __restrict__ row, const int* __restrict__ col,
                            const float* __restrict__ w, const float* __restrict__ dis,
                            const float* __restrict__ hw, float* __restrict__ agg,
                            int E) {
  unsigned t = blockIdx.x * blockDim.x + threadIdx.x;
  int e = (int)(t >> 4);
  if (e >= E) return;
  int cb = (int)(t & 15) << 2;
  int r = row[e];
  int c = col[e];
  float nrm = dis[r] * w[e] * dis[c];
  const float4 hv = *(const float4*)(hw + (size_t)r * HIDC + cb);
  float* ap = agg + (size_t)c * HIDC + cb;
  atomicAdd(ap + 0, hv.x * nrm);
  atomicAdd(ap + 1, hv.y * nrm);
  atomicAdd(ap + 2, hv.z * nrm);
  atomicAdd(ap + 3, hv.w * nrm);
}

// per-channel sum / sumsq into stat[0..63] / stat[64..127]
__global__ void gcn_bn_stats(const float* __restrict__ agg, float* __restrict__ stat,
                             int N) {
  int t = (int)(blockIdx.x * blockDim.x + threadIdx.x);
  int c = t & 63;
  int slot = t >> 6;
  int stride = (int)((gridDim.x * blockDim.x) >> 6);
  float s = 0.f, s2 = 0.f;
  for (int i = slot; i < N; i += stride) {
    float v = agg[(size_t)i * HIDC + c];
    s += v;
    s2 += v * v;
  }
  atomicAdd(&stat[c], s);
  atomicAdd(&stat[HIDC + c], s2);
}

// out = relu(g*(agg-mu)*rsqrt(var+eps)+beta) + out   (in-place residual add)
__global__ void gcn_bn_apply(const float* __restrict__ agg, const float* __restrict__ stat,
                             const float* __restrict__ g, const float* __restrict__ beta,
                             float* __restrict__ out, int N, float invN) {
  int idx = (int)(blockIdx.x * blockDim.x + threadIdx.x);
  if (idx < N * HIDC) {
    int c = idx & 63;
    float mu  = stat[c] * invN;
    float var = stat[HIDC + c] * invN - mu * mu;
    float v = g[c] * (agg[idx] - mu) * rsqrtf(var + BN_EPS) + beta[c];
    v = v > 0.0f ? v : 0.0f;
    out[idx] = v + out[idx];
  }
}

extern "C" void kernel_launch(void* const* d_in, const int* in_sizes, int n_in,
                              void* d_out, int out_size, void* d_ws, size_t ws_size,
                              hipStream_t stream) {
  const float* x   = (const float*)d_in[0];
  const int*   ei  = (const int*)d_in[1];
  const float* ew  = (const float*)d_in[2];
  const float* W1  = (const float*)d_in[3];
  const float* b1  = (const float*)d_in[4];
  const float* g1  = (const float*)d_in[5];
  const float* be1 = (const float*)d_in[6];
  const float* W2  = (const float*)d_in[7];
  const float* b2  = (const float*)d_in[8];
  const float* g2  = (const float*)d_in[9];
  const float* be2 = (const float*)d_in[10];
  const float* pW  = (const float*)d_in[11];
  const float* pb  = (const float*)d_in[12];
  float* out = (float*)d_out;

  const int HID  = in_sizes[4];            // 64
  const int INCH = in_sizes[3] / HID;      // 128
  const int N    = in_sizes[0] / INCH;     // 100000
  const int E    = in_sizes[2];            // 1600000
  const float invN = 1.0f / (float)N;

  const int* erow = ei;                    // edge_index[0]
  const int* ecol = ei + E;                // edge_index[1]

  // workspace layout: dis[N] | hw[N*64] | agg[N*64] | stat[128]
  float* dis  = (float*)d_ws;
  float* hw   = dis + N;
  float* agg  = hw + (size_t)N * HID;
  float* stat = agg + (size_t)N * HID;

  const int TB = 256;
  const int nWaves   = (N + 15) / 16;
  const int gemmBlk  = (nWaves * 32 + TB - 1) / TB;
  const int nodeBlk  = (N + TB - 1) / TB;
  const int nvBlk    = (N * HID + TB - 1) / TB;
  const int edgeBlk  = (E + TB - 1) / TB;
  const int scatBlk  = (int)(((size_t)E * 16 + TB - 1) / TB);

  // --- normalization: deg -> dis ---
  hipMemsetAsync(dis, 0, (size_t)N * sizeof(float), stream);
  gcn_deg<<<edgeBlk, TB, 0, stream>>>(ecol, ew, dis, E);
  gcn_dis<<<nodeBlk, TB, 0, stream>>>(dis, N);

  // --- residual = x @ pW + pb  (straight into d_out) ---
  gcn_gemm_wmma<<<gemmBlk, TB, 0, stream>>>(x, pW, pb, out, N, INCH);

  // --- layer 1 ---
  gcn_gemm_wmma<<<gemmBlk, TB, 0, stream>>>(x, W1, nullptr, hw, N, INCH);
  gcn_agg_init<<<nvBlk, TB, 0, stream>>>(hw, dis, b1, agg, N);
  gcn_scatter<<<scatBlk, TB, 0, stream>>>(erow, ecol, ew, dis, hw, agg, E);
  hipMemsetAsync(stat, 0, 2 * HID * sizeof(float), stream);
  gcn_bn_stats<<<256, TB, 0, stream>>>(agg, stat, N);
  gcn_bn_apply<<<nvBlk, TB, 0, stream>>>(agg, stat, g1, be1, out, N, invN);

  // --- layer 2 (A = d_out, K = 64) ---
  gcn_gemm_wmma<<<gemmBlk, TB, 0, stream>>>(out, W2, nullptr, hw, N, HID);
  gcn_agg_init<<<nvBlk, TB, 0, stream>>>(hw, dis, b2, agg, N);
  gcn_scatter<<<scatBlk, TB, 0, stream>>>(erow, ecol, ew, dis, hw, agg, E);
  hipMemsetAsync(stat, 0, 2 * HID * sizeof(float), stream);
  gcn_bn_stats<<<256, TB, 0, stream>>>(agg, stat, N);
  gcn_bn_apply<<<nvBlk, TB, 0, stream>>>(agg, stat, g2, be2, out, N, invN);
}